// LinearGaussianELBO_17480516894856
// MI455X (gfx1250) — compile-verified
//
#include <hip/hip_runtime.h>

#define D     32
#define LSTR  33          // padded LDS row stride (bank-conflict free)
#define TLEN  512
#define LOG2PI 1.8378770664093453f

typedef float v2f __attribute__((ext_vector_type(2)));
typedef float v8f __attribute__((ext_vector_type(8)));

// ---------------- workspace layout (float offsets) ----------------
constexpr int WS_BA   = 0;                       // bA[t]   t=1..511, 32x32
constexpr int WS_BCOV = WS_BA   + TLEN*D*D;      // bcov[t]
constexpr int WS_BAV  = WS_BCOV + TLEN*D*D;      // ba[t] vec
constexpr int WS_S    = WS_BAV  + TLEN*D;        // suffix S[t], t=1..512
constexpr int WS_V    = WS_S    + (TLEN+1)*D*D;  // suffix v[t]
constexpr int WS_PP   = WS_V    + (TLEN+1)*D;    // suffix Pprod[t]
constexpr int WS_FM   = WS_PP   + (TLEN+1)*D*D;  // final filter mean
constexpr int WS_FP   = WS_FM   + D;             // final filter cov
constexpr int WS_MEP  = WS_FP   + D*D;           // m_emis_prec
constexpr int WS_MTP  = WS_MEP  + D*D;           // m_trans_prec
constexpr int WS_MPP  = WS_MTP  + D*D;           // m_prior_prec
constexpr int WS_VTP  = WS_MPP  + D*D;           // v_trans_prec
constexpr int WS_COM  = WS_VTP  + D*D;           // common = Twᵀ Tprec
constexpr int WS_CW   = WS_COM  + D*D;           // common @ Tw
constexpr int WS_G    = WS_CW   + D*D;           // mTwᵀ mTprec mTw
constexpr int WS_CTB  = WS_G    + D*D;           // common @ v_trans_b
constexpr int WS_CB   = WS_CTB  + D;             // scalar constant
constexpr int WS_PART = WS_CB   + 1;             // per-slot partials (1024)
constexpr int WS_TOTAL= WS_PART + 2*TLEN;

// ---------------- wave helpers ----------------
__device__ __forceinline__ int laneid() { return threadIdx.x & 31; }

__device__ __forceinline__ float wred(float v) {
  v += __shfl_xor(v, 16, 32);
  v += __shfl_xor(v,  8, 32);
  v += __shfl_xor(v,  4, 32);
  v += __shfl_xor(v,  2, 32);
  v += __shfl_xor(v,  1, 32);
  return v;
}

// 32x32 C = opA(A) * opB(B) via V_WMMA_F32_16X16X4_F32, one full wave.
// Generic pointers: operands may live in LDS or global memory.
// Layouts per CDNA5 ISA 7.12.2:
//  A 16x4 : lanes0-15 M=l16, v0:K=0 v1:K=1 ; lanes16-31 M=l16, v0:K=2 v1:K=3
//  B 4x16 : v0: rows K=0|2 across lanes N=l16 ; v1: rows K=1|3
//  C 16x16: vgpr r -> M = r + 8*half, N = l16
template<bool TA, bool TB>
__device__ __forceinline__ void mm32(const float* A, int lda,
                                     const float* B, int ldb,
                                     float* C, int ldc) {
  const int lane = laneid();
  const int half = lane >> 4;
  const int l16  = lane & 15;
  #pragma unroll
  for (int mi = 0; mi < D; mi += 16) {
    #pragma unroll
    for (int nj = 0; nj < D; nj += 16) {
      v8f acc = {0.f,0.f,0.f,0.f,0.f,0.f,0.f,0.f};
      #pragma unroll
      for (int kk = 0; kk < D; kk += 4) {
        const int am = mi + l16;
        const int ak = kk + 2*half;
        const int bn = nj + l16;
        v2f af, bf;
        af[0] = TA ? A[(ak+0)*lda + am] : A[am*lda + (ak+0)];
        af[1] = TA ? A[(ak+1)*lda + am] : A[am*lda + (ak+1)];
        bf[0] = TB ? B[bn*ldb + (ak+0)] : B[(ak+0)*ldb + bn];
        bf[1] = TB ? B[bn*ldb + (ak+1)] : B[(ak+1)*ldb + bn];
        acc = __builtin_amdgcn_wmma_f32_16x16x4_f32(false, af, false, bf,
                                                    (short)0, acc, false, false);
      }
      #pragma unroll
      for (int r = 0; r < 8; ++r)
        C[(mi + r + 8*half)*ldc + (nj + l16)] = acc[r];
    }
  }
}

// y_i = sum_j M[i][j] x[j] ; lane i returns row-i dot.
__device__ __forceinline__ float matvec_row(const float* M, int ldm, const float* x) {
  const int lane = laneid();
  float s = 0.f;
  #pragma unroll 1
  for (int j = 0; j < D; ++j) s += M[lane*ldm + j] * x[j];
  return s;
}

// Gauss-Jordan inverse of SPD 32x32 in LDS (lane-per-column, single wave,
// no barriers needed: per-wave LDS ops are in order). Destroys W, writes V=W^-1,
// returns log|det(W)| from the pivot product.
__device__ __forceinline__ float gj_inverse(float* W, float* V) {
  const int lane = laneid();
  #pragma unroll 1
  for (int i = 0; i < D; ++i) V[i*LSTR + lane] = (i == lane) ? 1.f : 0.f;
  float ld = 0.f;
  #pragma unroll 1
  for (int k = 0; k < D; ++k) {
    const float piv = W[k*LSTR + k];
    ld += logf(fabsf(piv));
    const float ip = 1.f / piv;
    const float wk = W[k*LSTR + lane] * ip;
    const float vk = V[k*LSTR + lane] * ip;
    W[k*LSTR + lane] = wk;
    V[k*LSTR + lane] = vk;
    #pragma unroll 1
    for (int i = 0; i < D; ++i) {
      if (i == k) continue;
      const float m = W[i*LSTR + k];
      W[i*LSTR + lane] -= m * wk;
      V[i*LSTR + lane] -= m * vk;
    }
  }
  return ld;
}

__device__ __forceinline__ void g2l(const float* g, float* l) {
  const int lane = laneid();
  #pragma unroll 1
  for (int i = 0; i < D; ++i) l[i*LSTR + lane] = g[i*D + lane];
}
__device__ __forceinline__ void l2g(const float* l, float* g) {
  const int lane = laneid();
  #pragma unroll 1
  for (int i = 0; i < D; ++i) g[i*D + lane] = l[i*LSTR + lane];
}
__device__ __forceinline__ void l2l(const float* a, float* b) {
  const int lane = laneid();
  #pragma unroll 1
  for (int i = 0; i < D; ++i) b[i*LSTR + lane] = a[i*LSTR + lane];
}

__device__ __forceinline__ float cterm(float ld) {   // dim==32 everywhere
  return -0.5f * (32.0f * LOG2PI + ld);
}

// Kalman measurement update in place (fm vec LDS, fP mat LDS stride LSTR).
__device__ __forceinline__ void kf_update(float* fm, float* fP,
    const float* Ew, const float* Eb, const float* Ecov, const float* y,
    float* t2, float* W, float* invS, float* Kg, float* t3, float* rv) {
  const int lane = laneid();
  mm32<false,false>(Ew, D, fP, LSTR, t2, LSTR);            // t2 = Ew P
  mm32<false,true >(t2, LSTR, Ew, D, W, LSTR);             // W = Ew P Ewᵀ
  #pragma unroll 1
  for (int i = 0; i < D; ++i) W[i*LSTR + lane] += Ecov[i*D + lane];
  gj_inverse(W, invS);                                     // invS = S^-1
  mm32<true ,false>(t2, LSTR, invS, LSTR, Kg, LSTR);       // K = (EwP)ᵀ S^-1
  rv[lane] = y[lane] - (matvec_row(Ew, D, fm) + Eb[lane]);
  const float dm = matvec_row(Kg, LSTR, rv);
  mm32<false,false>(Kg, LSTR, t2, LSTR, t3, LSTR);         // t3 = K Ew P
  #pragma unroll 1
  for (int i = 0; i < D; ++i) fP[i*LSTR + lane] -= t3[i*LSTR + lane];
  fm[lane] += dm;
}

// Kalman time update in place.
__device__ __forceinline__ void kf_predict(float* fm, float* fP,
    const float* Tw, const float* Tb, const float* Tcov,
    float* t1, float* Pp) {
  const int lane = laneid();
  mm32<false,false>(Tw, D, fP, LSTR, t1, LSTR);            // t1 = Tw P
  mm32<false,true >(t1, LSTR, Tw, D, Pp, LSTR);            // Pp = Tw P Twᵀ
  #pragma unroll 1
  for (int i = 0; i < D; ++i) fP[i*LSTR + lane] = Pp[i*LSTR + lane] + Tcov[i*D + lane];
  const float nm = matvec_row(Tw, D, fm) + Tb[lane];
  fm[lane] = nm;
}

// =================== K1: forward Kalman pass (1 wave) ===================
__global__ void __launch_bounds__(32)
k1_forward(const float* obs,
           const float* m_prior_mean, const float* m_prior_cov,
           const float* m_trans_w, const float* m_trans_b, const float* m_trans_cov,
           const float* m_emis_w, const float* m_emis_b, const float* m_emis_cov,
           const float* v_prior_mean, const float* v_prior_cov,
           const float* v_trans_w, const float* v_trans_b, const float* v_trans_cov,
           const float* v_emis_w, const float* v_emis_b, const float* v_emis_cov,
           float* ws) {
  __shared__ float sm[8*LSTR*D];
  __shared__ float sv[3*D];
  float* s0 = sm + 0*LSTR*D;   // fP
  float* s1 = sm + 1*LSTR*D;   // GJ work
  float* s2 = sm + 2*LSTR*D;   // fprec / invS
  float* s3 = sm + 3*LSTR*D;   // bcov / K
  float* s4 = sm + 4*LSTR*D;   // bA / t3
  float* s5 = sm + 5*LSTR*D;   // t1
  float* s6 = sm + 6*LSTR*D;   // Pp
  float* s7 = sm + 7*LSTR*D;   // t2
  float* fm = sv;  float* v1 = sv + D;  float* rv = sv + 2*D;
  const int lane = laneid();

  // --- precisions & log-dets of model covariances ---
  g2l(m_prior_cov, s1); const float ld_prior = gj_inverse(s1, s2); l2g(s2, ws + WS_MPP);
  g2l(m_trans_cov, s1); const float ld_mt    = gj_inverse(s1, s2); l2g(s2, ws + WS_MTP);
  g2l(m_emis_cov , s1); const float ld_me    = gj_inverse(s1, s2); l2g(s2, ws + WS_MEP);
  g2l(v_trans_cov, s1);                        gj_inverse(s1, s2); l2g(s2, ws + WS_VTP);

  // --- step-invariant matrices ---
  mm32<true ,false>(v_trans_w, D, ws + WS_VTP, D, ws + WS_COM, D); // common = Twᵀ Tprec
  mm32<false,false>(ws + WS_COM, D, v_trans_w, D, ws + WS_CW, D);  // CW = common Tw
  mm32<false,false>(ws + WS_MTP, D, m_trans_w, D, s5, LSTR);       // tmp = mP mW
  mm32<true ,false>(m_trans_w, D, s5, LSTR, ws + WS_G, D);         // G = mWᵀ mP mW
  ws[WS_CTB + lane] = matvec_row(ws + WS_COM, D, v_trans_b);

  // --- t = 0 update ---
  fm[lane] = v_prior_mean[lane];
  g2l(v_prior_cov, s0);
  kf_update(fm, s0, v_emis_w, v_emis_b, v_emis_cov, obs, s7, s1, s2, s3, s4, rv);
  float cacc = cterm(ld_prior) + cterm(ld_me);

  // --- forward scan ---
  #pragma unroll 1
  for (int t = 1; t < TLEN; ++t) {
    l2l(s0, s1);
    gj_inverse(s1, s2);                               // fprec
    #pragma unroll 1
    for (int i = 0; i < D; ++i)
      s1[i*LSTR + lane] = ws[WS_CW + i*D + lane] + s2[i*LSTR + lane];
    const float ld_bcov = -gj_inverse(s1, s3);        // bcov = (CW+fprec)^-1
    mm32<false,false>(s3, LSTR, ws + WS_COM, D, s4, LSTR);  // bA = bcov common
    l2g(s4, ws + WS_BA   + t*D*D);
    l2g(s3, ws + WS_BCOV + t*D*D);
    v1[lane] = matvec_row(s2, LSTR, fm) - ws[WS_CTB + lane];
    ws[WS_BAV + t*D + lane] = matvec_row(s3, LSTR, v1);     // ba

    float fr = 0.f;                                   // tr(G bcov)
    #pragma unroll 1
    for (int i = 0; i < D; ++i) fr += ws[WS_G + i*D + lane] * s3[i*LSTR + lane];
    fr = wred(fr);
    cacc += cterm(ld_me) + cterm(ld_mt) - 0.5f*fr - cterm(ld_bcov) + 16.0f;

    kf_predict(fm, s0, v_trans_w, v_trans_b, v_trans_cov, s5, s6);
    kf_update(fm, s0, v_emis_w, v_emis_b, v_emis_cov, obs + t*D, s7, s1, s2, s3, s4, rv);
  }

  l2l(s0, s1);
  const float ld_fp = gj_inverse(s1, s2);
  cacc += -cterm(ld_fp) + 16.0f;
  ws[WS_FM + lane] = fm[lane];
  l2g(s0, ws + WS_FP);
  if (lane == 0) ws[WS_CB] = cacc;
}

// =================== K2: backward suffix pass (1 wave) ===================
__global__ void __launch_bounds__(32)
k2_backward(float* ws) {
  __shared__ float tmp[LSTR*D];
  const int lane = laneid();
  #pragma unroll 1
  for (int i = 0; i < D; ++i) {
    ws[WS_S  + TLEN*D*D + i*D + lane] = 0.f;
    ws[WS_PP + TLEN*D*D + i*D + lane] = (i == lane) ? 1.f : 0.f;
  }
  ws[WS_V + TLEN*D + lane] = 0.f;

  #pragma unroll 1
  for (int t = TLEN - 1; t >= 1; --t) {
    const float* bAt = ws + WS_BA + t*D*D;
    // S[t] = bcov[t] + bA S[t+1] bAᵀ
    mm32<false,false>(bAt, D, ws + WS_S + (t+1)*D*D, D, tmp, LSTR);
    mm32<false,true >(tmp, LSTR, bAt, D, ws + WS_S + t*D*D, D);
    #pragma unroll 1
    for (int i = 0; i < D; ++i)
      ws[WS_S + t*D*D + i*D + lane] += ws[WS_BCOV + t*D*D + i*D + lane];
    // v[t] = ba[t] + bA v[t+1]
    ws[WS_V + t*D + lane] =
        matvec_row(bAt, D, ws + WS_V + (t+1)*D) + ws[WS_BAV + t*D + lane];
    // Pprod[t] = bA Pprod[t+1]
    mm32<false,false>(bAt, D, ws + WS_PP + (t+1)*D*D, D, ws + WS_PP + t*D*D, D);
  }
}

// =================== K3: per-slot pass (1024 independent waves) ===================
__global__ void __launch_bounds__(64)
k3_slots(const float* obs, const float* m_prior_mean,
         const float* m_trans_w, const float* m_trans_b,
         const float* m_emis_w, const float* m_emis_b,
         float* ws) {
  __shared__ float sm[2][4*LSTR*D + 3*D];
  const int wave = threadIdx.x >> 5;
  const int lane = laneid();
  const int n  = blockIdx.x * 2 + wave;
  const int tn = n >> 1;
  const int u  = tn + 1;               // suffix index in [1, 512]
  float* A0 = sm[wave];
  float* M1 = A0 + 1*LSTR*D;
  float* M2 = A0 + 2*LSTR*D;
  float* M3 = A0 + 3*LSTR*D;
  float* bi = A0 + 4*LSTR*D;
  float* bf = bi + D;
  float* cm = bi + 2*D;

  __builtin_prefetch(&ws[WS_S  + u*D*D], 0, 1);
  __builtin_prefetch(&ws[WS_PP + u*D*D], 0, 1);

  const float* Ai; int ldai; const float* P;
  if ((n & 1) == 0) {                              // emission slot at t=tn
    Ai = m_emis_w; ldai = D;
    bi[lane] = m_emis_b[lane] - obs[tn*D + lane];
    P = ws + WS_MEP;
  } else if (n == 1) {                             // prior slot
    #pragma unroll 1
    for (int i = 0; i < D; ++i) A0[i*LSTR + lane] = (i == lane) ? 1.f : 0.f;
    Ai = A0; ldai = LSTR;
    bi[lane] = -m_prior_mean[lane];
    P = ws + WS_MPP;
  } else {                                         // transition slot at t=tn
    mm32<false,false>(m_trans_w, D, ws + WS_BA + tn*D*D, D, A0, LSTR);
    #pragma unroll 1
    for (int i = 0; i < D; ++i) A0[i*LSTR + lane] -= (i == lane) ? 1.f : 0.f;
    Ai = A0; ldai = LSTR;
    bi[lane] = matvec_row(m_trans_w, D, ws + WS_BAV + tn*D) + m_trans_b[lane];
    P = ws + WS_MTP;
  }

  // c1 = -0.5 tr( (Aᵀ P A) S[u] )
  mm32<false,false>(P, D, Ai, ldai, M1, LSTR);
  mm32<true ,false>(Ai, ldai, M1, LSTR, M2, LSTR);
  float fr = 0.f;
  #pragma unroll 1
  for (int i = 0; i < D; ++i)
    fr += M2[i*LSTR + lane] * ws[WS_S + u*D*D + i*D + lane];
  const float c1 = -0.5f * wred(fr);

  // Afin = A_init @ Pprod[u] ; bfin = b_init + A_init v[u]
  mm32<false,false>(Ai, ldai, ws + WS_PP + u*D*D, D, M1, LSTR);
  bf[lane] = bi[lane] + matvec_row(Ai, ldai, ws + WS_V + u*D);

  // tr2 = -0.5 tr( P · Afin fP Afin )   (note: A, not Aᵀ — faithful to source;
  // P symmetric, so tr(P·M2) = sum_ij P[i][j]*M2[i][j])
  mm32<false,false>(M1, LSTR, ws + WS_FP, D, M3, LSTR);
  mm32<false,false>(M3, LSTR, M1, LSTR, M2, LSTR);
  float fr2 = 0.f;
  #pragma unroll 1
  for (int i = 0; i < D; ++i)
    fr2 += P[i*D + lane] * M2[i*LSTR + lane];
  const float tr2 = -0.5f * wred(fr2);

  // qf = -0.5 cmᵀ P cm,  cm = Afin fm + bfin
  cm[lane] = matvec_row(M1, LSTR, ws + WS_FM) + bf[lane];
  const float yv = matvec_row(P, D, cm);
  const float qf = -0.5f * wred(cm[lane] * yv);

  if (lane == 0) ws[WS_PART + n] = c1 + tr2 + qf;
}

// =================== K4: deterministic reduction ===================
__global__ void __launch_bounds__(32)
k4_reduce(const float* ws, float* out) {
  const int lane = laneid();
  float s = 0.f;
  #pragma unroll 1
  for (int i = lane; i < 2*TLEN; i += 32) s += ws[WS_PART + i];
  s = wred(s);
  if (lane == 0) out[0] = s + ws[WS_CB];
}

extern "C" void kernel_launch(void* const* d_in, const int* in_sizes, int n_in,
                              void* d_out, int out_size, void* d_ws, size_t ws_size,
                              hipStream_t stream) {
  (void)in_sizes; (void)n_in; (void)out_size;
  if (ws_size < (size_t)WS_TOTAL * sizeof(float)) return;  // scratch guard
  const float* obs  = (const float*)d_in[0];
  const float* mpm  = (const float*)d_in[1];
  const float* mpc  = (const float*)d_in[2];
  const float* mtw  = (const float*)d_in[3];
  const float* mtb  = (const float*)d_in[4];
  const float* mtc  = (const float*)d_in[5];
  const float* mew  = (const float*)d_in[6];
  const float* meb  = (const float*)d_in[7];
  const float* mec  = (const float*)d_in[8];
  const float* vpm  = (const float*)d_in[9];
  const float* vpc  = (const float*)d_in[10];
  const float* vtw  = (const float*)d_in[11];
  const float* vtb  = (const float*)d_in[12];
  const float* vtc  = (const float*)d_in[13];
  const float* vew  = (const float*)d_in[14];
  const float* veb  = (const float*)d_in[15];
  const float* vec  = (const float*)d_in[16];
  float* ws  = (float*)d_ws;
  float* out = (float*)d_out;

  k1_forward<<<1, 32, 0, stream>>>(obs, mpm, mpc, mtw, mtb, mtc, mew, meb, mec,
                                   vpm, vpc, vtw, vtb, vtc, vew, veb, vec, ws);
  k2_backward<<<1, 32, 0, stream>>>(ws);
  k3_slots<<<TLEN, 64, 0, stream>>>(obs, mpm, mtw, mtb, mew, meb, ws);
  k4_reduce<<<1, 32, 0, stream>>>(ws, out);
}